// Model_80410377715763
// MI455X (gfx1250) — compile-verified
//
#include <hip/hip_runtime.h>

// MI455X / gfx1250 Radon transform via exact scatter-built banded GEMM + WMMA.
//
// out[b,0,a,d] = sum_t bilinear(img_b, xs(a,d,t), ys(a,d,t))
//   xs = (d-c)cos - (t-c)sin + c ;  ys = (d-c)sin + (t-c)cos + c ; c = 127.5
//
// Pass 1: transpose img [B][y][x] -> Xt [y][x][B]  (batch contiguous)
// Pass 2: per (angle, 16-detector tile): for each 16-ray chunk, scatter the
//         4 bilinear tap weights of every (d,t) into an LDS weight matrix
//         W[16][Kbox] over the chunk's pixel bounding box (ds_add_f32),
//         then D += W (16xK) * Xt_rows (Kx256 batches) with
//         v_wmma_f32_16x16x4_f32, K swept 4 at a time.

typedef float v2f __attribute__((ext_vector_type(2)));
typedef float v8f __attribute__((ext_vector_type(8)));

#define NPIX   256
#define NANG   256
#define NBATCH 256
#define TCHUNK 16
// worst-case padded bbox: Kx<=26 -> Kxp<=28, Ky<=26 -> 728; round up.
#define KMAX   736

// ---------------- Pass 1: [B][y][x] f32 -> [y][x][B] f32 ----------------
__global__ __launch_bounds__(256) void radon_transpose_kernel(
    const float* __restrict__ in, float* __restrict__ xt) {
  __shared__ float tile[32][33];
  const int y  = blockIdx.z;
  const int x0 = blockIdx.x * 32;
  const int b0 = blockIdx.y * 32;
  const int tid = threadIdx.x;
  const int l  = tid & 31;       // fast lane
  const int g  = tid >> 5;       // 0..7
#pragma unroll
  for (int ii = 0; ii < 4; ++ii) {
    const int bl = g + ii * 8;
    tile[bl][l] = in[(size_t)(b0 + bl) * (NPIX * NPIX) + (size_t)y * NPIX + x0 + l];
  }
  __syncthreads();
#pragma unroll
  for (int ii = 0; ii < 4; ++ii) {
    const int xl = g + ii * 8;
    xt[((size_t)y * NPIX + (x0 + xl)) * NBATCH + b0 + l] = tile[l][xl];
  }
}

// ---------------- Pass 2: banded WMMA radon ----------------
__global__ __launch_bounds__(256) void radon_wmma_kernel(
    const float* __restrict__ xt, const float* __restrict__ angles,
    float* __restrict__ out) {
  __shared__ float W[16 * KMAX];

  const int dtile = blockIdx.x;          // 0..15 (detector tile)
  const int a     = blockIdx.y;          // 0..255 (angle)
  const int d0    = dtile * 16;
  const int tid   = threadIdx.x;
  const int lane  = tid & 31;
  const int wv    = tid >> 5;            // wave 0..7
  const int lm    = lane & 15;           // M row (A) / N col (B,C)
  const int hi    = lane >> 4;           // lane-half -> K offset of +2

  const float theta = angles[a];
  const float cs = __cosf(theta);
  const float sn = __sinf(theta);
  const float cc = 127.5f;

  // two batch tiles per wave: columns [b0a, b0a+16) and [b0b, b0b+16)
  const int b0a = 32 * wv;
  const int b0b = b0a + 16;

  v8f c0 = {0.f, 0.f, 0.f, 0.f, 0.f, 0.f, 0.f, 0.f};
  v8f c1 = {0.f, 0.f, 0.f, 0.f, 0.f, 0.f, 0.f, 0.f};

  // scatter assignment: one (detector, ray) sample point per thread
  const int sdl = tid & 15;              // detector within tile
  const int stl = tid >> 4;              // ray within chunk

  for (int tc = 0; tc < NPIX / TCHUNK; ++tc) {
    const int t0 = tc * TCHUNK;

    // ---- uniform bounding box of all taps for (d in tile, t in chunk) ----
    const float dpl = (float)d0 - cc, dph = (float)(d0 + 15) - cc;
    const float tpl = (float)t0 - cc, tph = (float)(t0 + 15) - cc;
    const float xsA = dpl * cs - tpl * sn + cc;
    const float xsB = dpl * cs - tph * sn + cc;
    const float xsC = dph * cs - tpl * sn + cc;
    const float xsD = dph * cs - tph * sn + cc;
    const float ysA = dpl * sn + tpl * cs + cc;
    const float ysB = dpl * sn + tph * cs + cc;
    const float ysC = dph * sn + tpl * cs + cc;
    const float ysD = dph * sn + tph * cs + cc;
    const int xmin = (int)floorf(fminf(fminf(xsA, xsB), fminf(xsC, xsD))) - 1;
    const int xmax = (int)floorf(fmaxf(fmaxf(xsA, xsB), fmaxf(xsC, xsD))) + 2;
    const int ymin = (int)floorf(fminf(fminf(ysA, ysB), fminf(ysC, ysD))) - 1;
    const int ymax = (int)floorf(fmaxf(fmaxf(ysA, ysB), fmaxf(ysC, ysD))) + 2;
    const int Kx  = xmax - xmin + 1;           // <= 26
    const int Kxp = (Kx + 3) & ~3;             // padded row stride, mult of 4
    const int Ky  = ymax - ymin + 1;           // <= 26

    // ---- zero weight matrix ----
    for (int i = tid; i < 16 * KMAX; i += 256) W[i] = 0.f;
    __syncthreads();

    // ---- exact bilinear tap scatter (shared LDS float atomics) ----
    {
      const float dp = (float)(d0 + sdl) - cc;
      const float tp = (float)(t0 + stl) - cc;
      const float xs = dp * cs - tp * sn + cc;
      const float ys = dp * sn + tp * cs + cc;
      const float x0f = floorf(xs), y0f = floorf(ys);
      const float wx = xs - x0f, wy = ys - y0f;
      const int x0i = (int)x0f, y0i = (int)y0f;
      const float w00 = (1.f - wx) * (1.f - wy);
      const float w01 = wx * (1.f - wy);
      const float w10 = (1.f - wx) * wy;
      const float w11 = wx * wy;
      float* wr = &W[sdl * KMAX];
#define DEPOSIT(yy, xx, wt)                                                   \
      do {                                                                    \
        const int _y = (yy), _x = (xx);                                       \
        if (_y >= 0 && _y < NPIX && _x >= 0 && _x < NPIX)                     \
          atomicAdd(&wr[(_y - ymin) * Kxp + (_x - xmin)], (wt));              \
      } while (0)
      DEPOSIT(y0i,     x0i,     w00);
      DEPOSIT(y0i,     x0i + 1, w01);
      DEPOSIT(y0i + 1, x0i,     w10);
      DEPOSIT(y0i + 1, x0i + 1, w11);
#undef DEPOSIT
    }
    __syncthreads();

    // ---- WMMA GEMM over the bbox K-space (block-uniform loop bounds) ----
    for (int ky = 0; ky < Ky; ++ky) {
      const int py  = ymin + ky;
      const int pyc = min(max(py, 0), NPIX - 1);
      const float* __restrict__ xrow = xt + (size_t)pyc * (NPIX * NBATCH);
      const float* wrow = &W[lm * KMAX + ky * Kxp];
      for (int kx = 0; kx < Kxp; kx += 4) {
        const int kk = kx + 2 * hi;              // this lane-half's K pair
        // A fragment: W[lm][kk], W[lm][kk+1]  (8B-aligned LDS read)
        const v2f av = *(const v2f*)(wrow + kk);
        // B fragment: Xt[py][xmin+kk][b0+lm], Xt[py][xmin+kk+1][b0+lm]
        const int px0 = xmin + kk;
        const int p0 = min(max(px0, 0), NPIX - 1);
        const int p1 = min(max(px0 + 1, 0), NPIX - 1);
        v2f bv0, bv1;
        bv0.x = xrow[p0 * NBATCH + b0a + lm];
        bv0.y = xrow[p1 * NBATCH + b0a + lm];
        bv1.x = xrow[p0 * NBATCH + b0b + lm];
        bv1.y = xrow[p1 * NBATCH + b0b + lm];
        c0 = __builtin_amdgcn_wmma_f32_16x16x4_f32(false, av, false, bv0,
                                                   (short)0, c0, false, false);
        c1 = __builtin_amdgcn_wmma_f32_16x16x4_f32(false, av, false, bv1,
                                                   (short)0, c1, false, false);
      }
    }
    __syncthreads();
  }

  // ---- store C tiles: C[m][n] -> out[b0+n][0][a][d0+m] ----
#pragma unroll
  for (int r = 0; r < 8; ++r) {
    const int m = r + 8 * hi;
    const int d = d0 + m;
    out[(size_t)(b0a + lm) * (NANG * NPIX) + (size_t)a * NPIX + d] = c0[r];
    out[(size_t)(b0b + lm) * (NANG * NPIX) + (size_t)a * NPIX + d] = c1[r];
  }
}

// ---------------- Launch ----------------
extern "C" void kernel_launch(void* const* d_in, const int* in_sizes, int n_in,
                              void* d_out, int out_size, void* d_ws, size_t ws_size,
                              hipStream_t stream) {
  (void)in_sizes; (void)n_in; (void)out_size; (void)ws_size;
  const float* x      = (const float*)d_in[0];   // [256,1,256,256] f32
  const float* angles = (const float*)d_in[1];   // [256] f32
  float* out = (float*)d_out;                    // [256,1,256,256] f32
  float* xt  = (float*)d_ws;                     // 64 MiB scratch: Xt[y][x][b]

  dim3 tgrid(NPIX / 32, NBATCH / 32, NPIX);      // (x-tiles, b-tiles, y)
  radon_transpose_kernel<<<tgrid, 256, 0, stream>>>(x, xt);

  dim3 rgrid(NPIX / 16, NANG);                   // (detector tiles, angles)
  radon_wmma_kernel<<<rgrid, 256, 0, stream>>>(xt, angles, out);
}